// ToBEVHeightCompression_42279658062070
// MI455X (gfx1250) — compile-verified
//
#include <hip/hip_runtime.h>

// ToBEVHeightCompression for MI455X (gfx1250, wave32).
// Scatter-add N points x 32 channels into dense (2, 384, 400, 400) f32.
// Pure memory/atomic-bound op: no matmul content -> WMMA not applicable.
// CDNA5 paths: non-returning GLOBAL_ATOMIC_ADD_F32 (scope:SCOPE_DEV, STOREcnt),
// GLOBAL_STORE_B128 zero-fill, th:NT load for the read-once feats stream,
// S_LOAD_B128 (scalar constant cache) for wave-uniform coords.

#define CHANNELS 32
#define PS0 400   // bev dim 0 (x)
#define PS1 400   // bev dim 1 (y)
#define PS2 12    // height dim (z), clipped
// output: (BATCH, PS2*CHANNELS, PS0, PS1) = (2, 384, 400, 400)

typedef __attribute__((ext_vector_type(4))) float v4f;

// Non-returning f32 atomic add at device scope: global_atomic_add_f32,
// RMW executed at the L2 atomic units, no return-value latency.
__device__ __forceinline__ void atomic_add_f32_noret(float* p, float v) {
    asm volatile("global_atomic_add_f32 %0, %1, off scope:SCOPE_DEV"
                 :
                 : "v"(p), "v"(v)
                 : "memory");
}

__global__ void bev_zero_kernel(float* __restrict__ out, int n4) {
    int t = blockIdx.x * blockDim.x + threadIdx.x;
    if (t < n4) {
        v4f z = {0.0f, 0.0f, 0.0f, 0.0f};
        // Regular-temporal B128 store: keep zeroed lines resident in L2 so the
        // following atomic scatter hits them instead of fetching from HBM.
        ((v4f* __restrict__)out)[t] = z;
    }
}

__global__ void bev_scatter_kernel(const int* __restrict__ coords,
                                   const float* __restrict__ feats,
                                   float* __restrict__ out,
                                   int npts) {
    int t  = blockIdx.x * blockDim.x + threadIdx.x;
    int i  = t >> 5;        // point index: one wave32 per point
    int ch = t & 31;        // channel = lane id
    if (i >= npts) return;

    // i is uniform across the wave; make it provably scalar so the coords
    // fetch becomes a single s_load_b128 through the scalar constant cache
    // (KMcnt) instead of 32-lane broadcast VMEM loads.
    int iw = __builtin_amdgcn_readfirstlane(i);
    const int* c = coords + (size_t)iw * 4;
    int x = c[0];
    int z = c[1];
    int y = c[2];
    int b = c[3];
    z = min(max(z, 0), PS2 - 1);    // clip height dim

    // feats[i][ch]: contiguous 128B per wave, read exactly once -> NT hint
    // so this 38MB stream does not evict zeroed output lines from L2.
    float f = __builtin_nontemporal_load(feats + (size_t)iw * CHANNELS + ch);

    // dense idx folded through reshape(2,400,400,384)+transpose(0,3,1,2):
    // out[b][z*32+ch][x][y]
    int zc  = z * CHANNELS + ch;
    size_t idx = (((size_t)b * (PS2 * CHANNELS) + (size_t)zc) * PS0 + (size_t)x) * PS1
               + (size_t)y;

    atomic_add_f32_noret(out + idx, f);
}

extern "C" void kernel_launch(void* const* d_in, const int* in_sizes, int n_in,
                              void* d_out, int out_size, void* d_ws, size_t ws_size,
                              hipStream_t stream) {
    const int*   coords = (const int*)d_in[0];    // (N, 4) int32
    const float* feats  = (const float*)d_in[1];  // (N, 32) f32
    float*       out    = (float*)d_out;          // 2*384*400*400 f32

    int npts = in_sizes[0] / 4;

    // 1) Zero the dense output (122.88M floats) with B128 stores.
    int n4 = out_size >> 2;                       // out_size divisible by 4
    int zb = (n4 + 255) / 256;
    bev_zero_kernel<<<zb, 256, 0, stream>>>(out, n4);

    // 2) Scatter: one lane per (point, channel); 32 lanes = one point.
    long long total = (long long)npts * CHANNELS;
    int sb = (int)((total + 255) / 256);
    bev_scatter_kernel<<<sb, 256, 0, stream>>>(coords, feats, out, npts);
}